// GVCNN_21878563406288
// MI455X (gfx1250) — compile-verified
//
#include <hip/hip_runtime.h>
#include <hip/hip_bf16.h>

#define GV_N 8192
#define GV_V 12
#define GV_C 256
#define GV_EPS 1e-6f

typedef __attribute__((ext_vector_type(2))) float v2f;
typedef __attribute__((ext_vector_type(8))) float v8f;

// ---------------------------------------------------------------------------
// Kernel 1: per-(n,v) dot products via V_WMMA_F32_16X16X4_F32, then scores.
//
// Formulation: D(16x16) = A(16x4) * B(4x16), accumulated over 64 K-chunks.
//   A        = w-chunk replicated into ALL 16 rows (every lane loads the same
//              float2 of w; lanes<16 supply K=0,1, lanes>=16 supply K=2,3).
//              => every row of D equals the desired dots; no divergent loads.
//   B col j  = chunk of RPs row (rowbase + j): lane j loads a contiguous
//              float2 at RPs[row_j, k0 + 2*(lane>>4)].
//   D        = VGPR0, lanes 0-15  ->  M=0, N=j  (the 16 dots we consume).
// ---------------------------------------------------------------------------
__global__ __launch_bounds__(256) void gvcnn_scores_wmma(
    const float* __restrict__ RPs, const float* __restrict__ w,
    const float* __restrict__ bias, float* __restrict__ scores)
{
    const int lane = threadIdx.x & 31;
    const int wave = threadIdx.x >> 5;
    const int tile = blockIdx.x * 8 + wave;      // 16 flat (n,v) rows per tile
    const int row  = tile * 16 + (lane & 15);    // row this lane services
    const int koff = (lane >> 4) * 2;            // 0 for lanes<16, 2 for >=16

    const float* __restrict__ rowp = RPs + (size_t)row * GV_C + koff;
    const float* __restrict__ wp   = w + koff;

    v8f acc = {};
    #pragma unroll 8
    for (int k0 = 0; k0 < GV_C; k0 += 4) {
        v2f bmat = *(const v2f*)(rowp + k0);     // contiguous 8B per lane
        v2f amat = *(const v2f*)(wp + k0);       // uniform w chunk, L0-cached
        acc = __builtin_amdgcn_wmma_f32_16x16x4_f32(
            /*neg_a=*/false, amat, /*neg_b=*/false, bmat,
            /*c_mod=*/(short)0, acc, /*reuse_a=*/false, /*reuse_b=*/false);
    }

    if (lane < 16) {
        float y = acc[0] + bias[0];
        // sigmoid(log(|y|+eps)) == t/(1+t) with t = |y|+eps   (exact identity)
        float t = fabsf(y) + GV_EPS;
        scores[row] = t / (t + 1.0f);
    }
}

// ---------------------------------------------------------------------------
// Kernel 2: per-n group pooling collapsed to a per-view scalar weight.
//   gid[v]        = min(int(score*10),9) >> 1
//   gsize[g]      = #views in bin g
//   group_score[g]= sum_{v in g} ceil(score_v * gsize_g) / (gsize_g + eps)
//   alpha[v]      = group_score[gid_v] / (gsize[gid_v] + eps)
//   out[n,c]      = sum_v alpha[v]*RPs[n,v,c] / (sum_g group_score + eps)
// One block per n; thread 0 does the 12-element scalar math, 256 threads do
// the coalesced weighted feature sum (RPs re-read hits the 192MB L2).
// ---------------------------------------------------------------------------
__global__ __launch_bounds__(256) void gvcnn_pool(
    const float* __restrict__ RPs, const float* __restrict__ scores,
    float* __restrict__ out)
{
    const int n = blockIdx.x;
    __shared__ float s_alpha[GV_V];
    __shared__ float s_inv_denom;

    if (threadIdx.x == 0) {
        float sc[GV_V];
        int   gid[GV_V];
        float gsize[5]  = {0.f, 0.f, 0.f, 0.f, 0.f};
        float gscore[5] = {0.f, 0.f, 0.f, 0.f, 0.f};
        #pragma unroll
        for (int v = 0; v < GV_V; ++v) {
            float s = scores[n * GV_V + v];
            int   g = min((int)(s * 10.0f), 9) >> 1;
            sc[v] = s; gid[v] = g;
            gsize[g] += 1.0f;
        }
        #pragma unroll
        for (int v = 0; v < GV_V; ++v)
            gscore[gid[v]] += ceilf(sc[v] * gsize[gid[v]]);
        float S = 0.f;
        #pragma unroll
        for (int g = 0; g < 5; ++g) {
            gscore[g] = gscore[g] / (gsize[g] + GV_EPS);   // == group_score
            S += gscore[g];
        }
        #pragma unroll
        for (int v = 0; v < GV_V; ++v)
            s_alpha[v] = gscore[gid[v]] / (gsize[gid[v]] + GV_EPS);
        s_inv_denom = 1.0f / (S + GV_EPS);
    }
    __syncthreads();

    const int c = threadIdx.x;
    const float* __restrict__ base = RPs + (size_t)n * GV_V * GV_C + c;
    float a = 0.f;
    #pragma unroll
    for (int v = 0; v < GV_V; ++v)
        a += s_alpha[v] * base[v * GV_C];
    out[n * GV_C + c] = a * s_inv_denom;
}

extern "C" void kernel_launch(void* const* d_in, const int* in_sizes, int n_in,
                              void* d_out, int out_size, void* d_ws, size_t ws_size,
                              hipStream_t stream) {
    const float* RPs  = (const float*)d_in[0];   // [N, V, C] f32
    const float* w    = (const float*)d_in[1];   // [C] f32
    const float* bias = (const float*)d_in[2];   // [1] f32
    float* out    = (float*)d_out;               // [N, C] f32
    float* scores = (float*)d_ws;                // N*V floats = 384 KB scratch

    // 98304 rows / 16 rows-per-wave / 8 waves-per-block = 768 blocks
    gvcnn_scores_wmma<<<(GV_N * GV_V) / (16 * 8), 256, 0, stream>>>(
        RPs, w, bias, scores);
    gvcnn_pool<<<GV_N, 256, 0, stream>>>(RPs, scores, out);
}